// RBF_Kernel_Layer_78469052498128
// MI455X (gfx1250) — compile-verified
//
#include <hip/hip_runtime.h>

typedef __attribute__((ext_vector_type(2))) float v2f;
typedef __attribute__((ext_vector_type(8))) float v8f;

#define B_  8
#define T_  512
#define D_  64

// ---------------------------------------------------------------------------
// Kernel 1: v[m,e] = sum_d x[m,d] * W[e,d] + b[e]   (m = b*T + t, 4096 x 64)
// Full-precision fp32 WMMA: V_WMMA_F32_16X16X4_F32, one wave per 16x16 tile.
//
// Fragment layouts (CDNA5 ISA 7.12.2, 32-bit data, wave32):
//   A (16x4, MxK):  lanes 0-15: M=lane,    VGPR0=K0, VGPR1=K1
//                   lanes16-31: M=lane-16, VGPR0=K2, VGPR1=K3
//   B (4x16, KxN):  mirrored (rows striped across lanes within a VGPR):
//                   lanes 0-15: N=lane,    VGPR0=K0, VGPR1=K1
//                   lanes16-31: N=lane-16, VGPR0=K2, VGPR1=K3
//     B[k][n] = W[n][k], so the B-frag indexes W rows exactly like A indexes x.
//   C/D (16x16):    VGPR r: lanes 0-15 -> M=r, lanes 16-31 -> M=r+8, N=lane&15
// ---------------------------------------------------------------------------
__global__ __launch_bounds__(32) void linear_wmma(const float* __restrict__ x,
                                                  const float* __restrict__ W,
                                                  const float* __restrict__ bias,
                                                  float* __restrict__ v) {
  const int tile = blockIdx.x;          // 1024 tiles = 256 m-tiles * 4 e-tiles
  const int m0   = (tile >> 2) * 16;
  const int e0   = (tile & 3) * 16;
  const int lane = threadIdx.x;
  const int half = lane >> 4;           // 0: K0..1 / M=r ; 1: K2..3 / M=r+8
  const int l    = lane & 15;

  const float* xrow = x + (size_t)(m0 + l) * D_;   // A-matrix row (M = m0+l)
  const float* wrow = W + (size_t)(e0 + l) * D_;   // B-matrix col (N = e0+l)

  v8f c = {};
#pragma unroll
  for (int k0 = 0; k0 < D_; k0 += 4) {
    v2f a, bf;
    a.x  = xrow[k0 + 2 * half];
    a.y  = xrow[k0 + 2 * half + 1];
    bf.x = wrow[k0 + 2 * half];
    bf.y = wrow[k0 + 2 * half + 1];
    // 8 args: (neg_a, A, neg_b, B, c_mod, C, reuse_a, reuse_b)
    c = __builtin_amdgcn_wmma_f32_16x16x4_f32(false, a, false, bf,
                                              (short)0, c, false, false);
  }

  const float bb = bias[e0 + l];
#pragma unroll
  for (int r = 0; r < 8; ++r) {
    const int m = m0 + r + 8 * half;
    v[(size_t)m * D_ + (e0 + l)] = c[r] + bb;
  }
}

// ---------------------------------------------------------------------------
// Kernel 2: out[b,i,j] = sum_d e(i,j,d)*v[b,i,d] / sum_d e(i,j,d),
//           e = exp(-(x[b,i,d]-x[b,j,d])^2).
// Block = (b, 8 i-rows), 128 threads; each thread owns one j per 128-j chunk.
// xi/vi rows + the j-chunk staged in LDS (stride-65 pad => conflict-free:
// bank(tid*65+d) = (tid+d) mod 64). 8 independent accumulator pairs per
// thread keep the v_exp_f32 (trans) pipe saturated -- the workload's limiter.
// exp without max-shift is safe: args <= 0, exp in (0,1].
// ---------------------------------------------------------------------------
#define ITILE  8
#define JCHUNK 128
#define JPAD   65

__global__ __launch_bounds__(128) void rbf_pairwise(const float* __restrict__ x,
                                                    const float* __restrict__ v,
                                                    float* __restrict__ out) {
  __shared__ float xi[ITILE][D_];
  __shared__ float vi[ITILE][D_];
  __shared__ float xj[JCHUNK * JPAD];

  const int blk = blockIdx.x;                  // B * (T/ITILE) = 512 blocks
  const int b   = blk / (T_ / ITILE);
  const int i0  = (blk % (T_ / ITILE)) * ITILE;
  const int tid = threadIdx.x;

  // Stage the 8 query rows of x and v (512 floats each).
  for (int t = tid; t < ITILE * D_; t += 128) {
    const int ii = t >> 6, d = t & 63;
    const size_t g = (size_t)(b * T_ + i0 + ii) * D_ + d;
    xi[ii][d] = x[g];
    vi[ii][d] = v[g];
  }

  for (int j0 = 0; j0 < T_; j0 += JCHUNK) {
    __syncthreads();  // prev chunk consumed (and xi/vi visible on first iter)
    // Stage x[b, j0 .. j0+127, :] -- coalesced global reads.
    for (int t = tid; t < JCHUNK * D_; t += 128) {
      const int jj = t >> 6, d = t & 63;
      xj[jj * JPAD + d] = x[(size_t)(b * T_ + j0 + jj) * D_ + d];
    }
    __syncthreads();

    const int j = j0 + tid;
    float num[ITILE], den[ITILE];
#pragma unroll
    for (int ii = 0; ii < ITILE; ++ii) { num[ii] = 0.0f; den[ii] = 0.0f; }

    const float* xjrow = &xj[tid * JPAD];
#pragma unroll 4
    for (int d = 0; d < D_; ++d) {
      const float xjd = xjrow[d];
#pragma unroll
      for (int ii = 0; ii < ITILE; ++ii) {
        const float diff = xi[ii][d] - xjd;
        // exp(-diff^2) = exp2(-diff^2 * log2(e)) -> single v_exp_f32
        const float e = __builtin_amdgcn_exp2f(diff * diff * -1.4426950408889634f);
        den[ii] += e;
        num[ii] += e * vi[ii][d];
      }
    }

#pragma unroll
    for (int ii = 0; ii < ITILE; ++ii) {
      out[(size_t)(b * T_ + i0 + ii) * T_ + j] = num[ii] / den[ii];
    }
  }
}

extern "C" void kernel_launch(void* const* d_in, const int* in_sizes, int n_in,
                              void* d_out, int out_size, void* d_ws, size_t ws_size,
                              hipStream_t stream) {
  const float* x    = (const float*)d_in[0];   // (B,T,D) fp32
  const float* W    = (const float*)d_in[1];   // (D,D)   fp32
  const float* bias = (const float*)d_in[2];   // (D,)    fp32
  float*       out  = (float*)d_out;           // (B,T,T) fp32
  float*       v    = (float*)d_ws;            // (B*T, D) fp32 scratch = 1 MB

  // 256 m-tiles * 4 e-tiles, one wave32 per 16x16 tile.
  linear_wmma<<<dim3((B_ * T_ / 16) * (D_ / 16)), dim3(32), 0, stream>>>(x, W, bias, v);
  // 512 blocks of (b, 8 i-rows).
  rbf_pairwise<<<dim3(B_ * (T_ / ITILE)), dim3(128), 0, stream>>>(x, v, out);
}